// Lut3D_15315853377720
// MI455X (gfx1250) — compile-verified
//
#include <hip/hip_runtime.h>
#include <cstdint>
#include <cstddef>

// ---------------------------------------------------------------------------
// 3D LUT (33^3, 3-channel) trilinear interpolation over 16x3x1080x1920 fp32.
//
// Roofline (MI455X): 796 MB streaming @ 23.3 TB/s -> ~34 us; plus ~4.25 GB of
// divergent LUT gathers served from the 192 MB L2 (LUT = 575 KB repacked,
// permanently resident). Gather + pointwise -> WMMA inapplicable by design.
//
// CDNA5-specific paths used:
//  * GLOBAL_LOAD_ASYNC_TO_LDS_B128 (ASYNCcnt) double-buffered pipeline for the
//    streaming input: input for chunk c+1 is in flight while chunk c's
//    data-dependent LUT gathers + lerps execute.
//  * s_wait_asynccnt split-counter sync (per-wave, no barrier needed since each
//    lane consumes only its own LDS slot).
//  * TH_NT non-temporal stores on the output stream so it cannot evict the
//    L2-resident LUT.
// ---------------------------------------------------------------------------

typedef __attribute__((ext_vector_type(4))) float v4f;
typedef __attribute__((ext_vector_type(2), aligned(4))) float v2fu;

#define DIM    33
#define DIM2   (DIM * DIM)            // 1089
#define NCELL  (DIM * DIM * DIM)      // 35937
#define PLANE  (1080 * 1920)          // 2073600 (divisible by 4)
#define NIMG   16
#define NPIX   ((long long)NIMG * (long long)PLANE)  // 33,177,600
#define TPB    256
#define CHUNKS 4                       // 4 px/thread/chunk -> 4096 px/block

__device__ __forceinline__ v4f lerp4(v4f a, v4f b, float t) {
    return a + t * (b - a);
}
__device__ __forceinline__ float lerp1(float a, float b, float t) {
    return a + t * (b - a);
}

// --- Pass 1: repack planar lut[c][b][g][r] -> interleaved float4 cells ------
__global__ __launch_bounds__(256) void repack_lut_kernel(
    const float* __restrict__ lut, v4f* __restrict__ lutc) {
    int i = blockIdx.x * blockDim.x + threadIdx.x;
    if (i < NCELL) {
        v4f c;
        c.x = lut[i];
        c.y = lut[NCELL + i];
        c.z = lut[2 * NCELL + i];
        c.w = 0.0f;
        lutc[i] = c;   // store warms L2 with the LUT working set
    }
}

// 8 b128 gathers + 7 float4 lerps for a group of (already loaded) r,g,b.
__device__ __forceinline__ void interp4(
    const v4f& rv, const v4f& gv, const v4f& bv,
    const v4f* __restrict__ lutc, v4f& orv, v4f& ogv, v4f& obv) {
    const float scale = 32.0f / 1.000001f;   // 1/binsize, dim-1 = 32
#pragma unroll
    for (int k = 0; k < 4; ++k) {
        float tr = rv[k] * scale;
        float tg = gv[k] * scale;
        float tb = bv[k] * scale;
        float fr = floorf(tr), fg = floorf(tg), fb = floorf(tb);
        float rd = tr - fr, gd = tg - fg, bd = tb - fb;  // frac before clip (matches ref)
        int r0 = min(max((int)fr, 0), DIM - 2);
        int g0 = min(max((int)fg, 0), DIM - 2);
        int b0 = min(max((int)fb, 0), DIM - 2);
        int base = (b0 * DIM + g0) * DIM + r0;

        v4f v000 = lutc[base],              v100 = lutc[base + 1];
        v4f v010 = lutc[base + DIM],        v110 = lutc[base + DIM + 1];
        v4f v001 = lutc[base + DIM2],       v101 = lutc[base + DIM2 + 1];
        v4f v011 = lutc[base + DIM2 + DIM], v111 = lutc[base + DIM2 + DIM + 1];

        v4f c00 = lerp4(v000, v100, rd);
        v4f c10 = lerp4(v010, v110, rd);
        v4f c01 = lerp4(v001, v101, rd);
        v4f c11 = lerp4(v011, v111, rd);
        v4f c0  = lerp4(c00, c10, gd);
        v4f c1  = lerp4(c01, c11, gd);
        v4f c   = lerp4(c0, c1, bd);

        orv[k] = c.x; ogv[k] = c.y; obv[k] = c.z;
    }
}

// --- Pass 2: async-LDS double-buffered pipeline over the streaming input ----
__global__ __launch_bounds__(TPB) void lut3d_apply_pipelined(
    const float* __restrict__ x, const v4f* __restrict__ lutc,
    float* __restrict__ out) {
    // 2 buffers x 3 planes x 256 lanes x 16 B = 24 KB LDS
    __shared__ v4f sbuf[2][3][TPB];

    const unsigned tid = threadIdx.x;
    const size_t   g0  = (size_t)blockIdx.x * (CHUNKS * TPB) + tid; // 4-px group idx

    // Issue chunk c's three async global->LDS B128 loads (per-lane slots).
    auto issue = [&](int c) {
        size_t g   = g0 + (size_t)c * TPB;
        size_t p   = g * 4;                       // pixel index (PLANE % 4 == 0)
        size_t n   = p / (size_t)PLANE;
        size_t rem = p - n * (size_t)PLANE;
        const float* xb = x + n * (size_t)(3 * PLANE) + rem;
        // low 32 bits of a generic pointer to LDS == wave-relative LDS byte addr
        unsigned l0 = (unsigned)(size_t)&sbuf[c & 1][0][tid];
        unsigned l1 = (unsigned)(size_t)&sbuf[c & 1][1][tid];
        unsigned l2 = (unsigned)(size_t)&sbuf[c & 1][2][tid];
        asm volatile("global_load_async_to_lds_b128 %0, %1, off"
                     :: "v"(l0), "v"(xb) : "memory");
        asm volatile("global_load_async_to_lds_b128 %0, %1, off"
                     :: "v"(l1), "v"(xb + PLANE) : "memory");
        asm volatile("global_load_async_to_lds_b128 %0, %1, off"
                     :: "v"(l2), "v"(xb + 2 * PLANE) : "memory");
    };

    issue(0);
#pragma unroll
    for (int c = 0; c < CHUNKS; ++c) {
        if (c + 1 < CHUNKS) {
            issue(c + 1);
            // 3 newly issued may stay outstanding; chunk c's 3 are done
            // (async loads complete in order).
            asm volatile("s_wait_asynccnt 0x3" ::: "memory");
        } else {
            asm volatile("s_wait_asynccnt 0x0" ::: "memory");
        }

        // Per-lane LDS slots -> no workgroup barrier required.
        v4f rv = sbuf[c & 1][0][tid];
        v4f gv = sbuf[c & 1][1][tid];
        v4f bv = sbuf[c & 1][2][tid];

        v4f orv, ogv, obv;
        interp4(rv, gv, bv, lutc, orv, ogv, obv);

        size_t g   = g0 + (size_t)c * TPB;
        size_t p   = g * 4;
        size_t n   = p / (size_t)PLANE;
        size_t rem = p - n * (size_t)PLANE;
        float* ob  = out + n * (size_t)(3 * PLANE) + rem;
        // Non-temporal stores: keep the output stream out of the LUT's L2 set.
        __builtin_nontemporal_store(orv, (v4f*)(ob));
        __builtin_nontemporal_store(ogv, (v4f*)(ob + PLANE));
        __builtin_nontemporal_store(obv, (v4f*)(ob + 2 * PLANE));
    }
}

// --- Fallback: direct planar-LUT path (used only if d_ws is too small) ------
__global__ __launch_bounds__(256) void lut3d_apply_direct(
    const float* __restrict__ x, const float* __restrict__ lut,
    float* __restrict__ out) {
    const float scale = 32.0f / 1.000001f;

    size_t tid = (size_t)blockIdx.x * blockDim.x + threadIdx.x;
    size_t p   = tid * 4;
    if (p >= (size_t)NPIX) return;

    size_t n   = p / (size_t)PLANE;
    size_t rem = p - n * (size_t)PLANE;

    const float* xb = x + n * (size_t)(3 * PLANE) + rem;
    v4f rv = __builtin_nontemporal_load((const v4f*)(xb));
    v4f gv = __builtin_nontemporal_load((const v4f*)(xb + PLANE));
    v4f bv = __builtin_nontemporal_load((const v4f*)(xb + 2 * PLANE));

    v4f outc[3];
#pragma unroll
    for (int k = 0; k < 4; ++k) {
        float tr = rv[k] * scale;
        float tg = gv[k] * scale;
        float tb = bv[k] * scale;
        float fr = floorf(tr), fg = floorf(tg), fb = floorf(tb);
        float rd = tr - fr, gd = tg - fg, bd = tb - fb;
        int r0 = min(max((int)fr, 0), DIM - 2);
        int g0 = min(max((int)fg, 0), DIM - 2);
        int b0 = min(max((int)fb, 0), DIM - 2);
        int base = (b0 * DIM + g0) * DIM + r0;

#pragma unroll
        for (int c = 0; c < 3; ++c) {
            const float* lp = lut + (size_t)c * NCELL + base;
            v2fu p00 = *(const v2fu*)(lp);
            v2fu p10 = *(const v2fu*)(lp + DIM);
            v2fu p01 = *(const v2fu*)(lp + DIM2);
            v2fu p11 = *(const v2fu*)(lp + DIM2 + DIM);
            float c00 = lerp1(p00.x, p00.y, rd);
            float c10 = lerp1(p10.x, p10.y, rd);
            float c01 = lerp1(p01.x, p01.y, rd);
            float c11 = lerp1(p11.x, p11.y, rd);
            float c0  = lerp1(c00, c10, gd);
            float c1  = lerp1(c01, c11, gd);
            outc[c][k] = lerp1(c0, c1, bd);
        }
    }

    float* ob = out + n * (size_t)(3 * PLANE) + rem;
    __builtin_nontemporal_store(outc[0], (v4f*)(ob));
    __builtin_nontemporal_store(outc[1], (v4f*)(ob + PLANE));
    __builtin_nontemporal_store(outc[2], (v4f*)(ob + 2 * PLANE));
}

extern "C" void kernel_launch(void* const* d_in, const int* in_sizes, int n_in,
                              void* d_out, int out_size, void* d_ws, size_t ws_size,
                              hipStream_t stream) {
    (void)in_sizes; (void)n_in; (void)out_size;
    const float* lut = (const float*)d_in[0];   // (3, 33, 33, 33) fp32
    const float* x   = (const float*)d_in[1];   // (16, 3, 1080, 1920) fp32
    float*       out = (float*)d_out;           // (16, 3, 1080, 1920) fp32

    const size_t need = (size_t)NCELL * sizeof(v4f);   // 574,992 B

    if (d_ws != nullptr && ws_size >= need) {
        v4f* lutc = (v4f*)d_ws;
        repack_lut_kernel<<<(NCELL + 255) / 256, 256, 0, stream>>>(lut, lutc);
        // NPIX / (TPB * CHUNKS * 4 px) = 33,177,600 / 4096 = 8100 blocks exactly
        const int blocks = (int)(NPIX / ((long long)TPB * CHUNKS * 4));
        lut3d_apply_pipelined<<<blocks, TPB, 0, stream>>>(x, lutc, out);
    } else {
        const long long nthreads = NPIX / 4;
        const int blocks = (int)((nthreads + 255) / 256);
        lut3d_apply_direct<<<blocks, 256, 0, stream>>>(x, lut, out);
    }
}